// HCHA_31842887533239
// MI455X (gfx1250) — compile-verified
//
#include <hip/hip_runtime.h>

#define IN_C  128
#define HEADS 4
#define FH    32

typedef __attribute__((ext_vector_type(2))) float v2f;
typedef __attribute__((ext_vector_type(8))) float v8f;

// ---- order-preserving float <-> uint keys (for atomicMax on floats) ----
__device__ __forceinline__ unsigned f32key(float f) {
  unsigned u = __float_as_uint(f);
  return (u & 0x80000000u) ? ~u : (u | 0x80000000u);
}
__device__ __forceinline__ float keyf32(unsigned k) {
  return __uint_as_float((k & 0x80000000u) ? (k & 0x7fffffffu) : ~k);
}

// ---------------- fills ----------------
__global__ void fill_f32_k(float* p, float v, long long n) {
  long long i = (long long)blockIdx.x * blockDim.x + threadIdx.x;
  long long s = (long long)gridDim.x * blockDim.x;
  for (; i < n; i += s) p[i] = v;
}
__global__ void fill_u32_k(unsigned* p, unsigned v, long long n) {
  long long i = (long long)blockIdx.x * blockDim.x + threadIdx.x;
  long long s = (long long)gridDim.x * blockDim.x;
  for (; i < n; i += s) p[i] = v;
}

// ---------------- degrees ----------------
__global__ void deg_k(const int* __restrict__ ni, const int* __restrict__ ei,
                      float* deg, float* bsz, int nnz) {
  int i = blockIdx.x * blockDim.x + threadIdx.x;
  if (i < nnz) {
    atomicAdd(&deg[ni[i]], 1.0f);
    atomicAdd(&bsz[ei[i]], 1.0f);
  }
}
__global__ void recip_k(float* p, int n) {
  int i = blockIdx.x * blockDim.x + threadIdx.x;
  if (i < n) { float v = p[i]; p[i] = v > 0.f ? 1.f / v : 0.f; }
}

// ---------------- WMMA GEMM: C[rows x 128] = A[rows x 128] * W^T ----------------
// W is [128 x 128] row-major (out x in). One wave computes a 16-row x 128-col strip
// using V_WMMA_F32_16X16X4_F32, K swept in steps of 4 (32 WMMAs per col tile).
__global__ __launch_bounds__(256) void gemm_nt_wmma(const float* __restrict__ A,
                                                    const float* __restrict__ W,
                                                    float* __restrict__ C, int rows) {
  int wave = blockIdx.x * (blockDim.x >> 5) + (threadIdx.x >> 5);
  int lane = threadIdx.x & 31;
  int rowBase = wave << 4;
  if (rowBase >= rows) return;          // wave-uniform: EXEC stays all-ones
  int lg = lane >> 4;                   // lane group (K offset 0 or 2)
  int lr = lane & 15;                   // row (A) / col (B) within tile

  v8f acc[8] = {};
  const float* arow = A + (size_t)(rowBase + lr) * IN_C;

#pragma unroll 4
  for (int k = 0; k < IN_C; k += 4) {
    int kk = k + (lg << 1);
    v2f a;
    a.x = arow[kk];
    a.y = arow[kk + 1];
#pragma unroll
    for (int t = 0; t < 8; ++t) {
      const float* wp = W + (size_t)(t * 16 + lr) * IN_C + kk;  // B[k][col]=W[col][k]
      v2f b;
      b.x = wp[0];
      b.y = wp[1];
      acc[t] = __builtin_amdgcn_wmma_f32_16x16x4_f32(
          /*neg_a=*/false, a, /*neg_b=*/false, b,
          /*c_mod=*/(short)0, acc[t], /*reuse_a=*/false, /*reuse_b=*/false);
    }
  }
  // C/D layout: VGPR r -> M=r (lanes 0-15) / M=r+8 (lanes 16-31), N=lr
#pragma unroll
  for (int t = 0; t < 8; ++t) {
    int col = t * 16 + lr;
#pragma unroll
    for (int r = 0; r < 8; ++r) {
      int row = rowBase + r + (lg << 3);
      C[(size_t)row * IN_C + col] = acc[t][r];
    }
  }
}

// ---------------- attention: logits + per-(node,head) max ----------------
__global__ __launch_bounds__(256) void attn_logits_k(
    const float* __restrict__ x1, const float* __restrict__ he,
    const float* __restrict__ att1, const int* __restrict__ ni,
    const int* __restrict__ ei, float* __restrict__ logits,
    unsigned* __restrict__ nmax, int nnz) {
  long long gtid = (long long)blockIdx.x * blockDim.x + threadIdx.x;
  int lane = (int)(gtid & 31);
  long long w = gtid >> 5;      // one wave per (nnz, head)
  long long i = w >> 2;
  int head = (int)(w & 3);
  if (i >= nnz) return;
  int node = ni[i], edge = ei[i];
  int f = head * FH + lane;
  float v = x1[(size_t)node * IN_C + f] * att1[head * 2 * FH + lane] +
            he[(size_t)edge * IN_C + f] * att1[head * 2 * FH + FH + lane];
#pragma unroll
  for (int off = 16; off > 0; off >>= 1) v += __shfl_down(v, off, 32);
  if (lane == 0) {
    float l = v > 0.f ? v : 0.2f * v;   // leaky_relu
    logits[i * 4 + head] = l;
    atomicMax(&nmax[(size_t)node * 4 + head], f32key(l));
  }
}

__global__ void attn_exp_k(const int* __restrict__ ni, const unsigned* __restrict__ nmax,
                           float* __restrict__ logits, float* __restrict__ nsum,
                           long long nnz4) {
  long long idx = (long long)blockIdx.x * blockDim.x + threadIdx.x;
  if (idx >= nnz4) return;
  long long i = idx >> 2;
  int h = (int)(idx & 3);
  int node = ni[i];
  float m = keyf32(nmax[(size_t)node * 4 + h]);
  float e = expf(logits[idx] - m);
  logits[idx] = e;
  atomicAdd(&nsum[(size_t)node * 4 + h], e);
}

__global__ void attn_norm_k(const int* __restrict__ ni, const float* __restrict__ nsum,
                            float* __restrict__ alpha, long long nnz4) {
  long long idx = (long long)blockIdx.x * blockDim.x + threadIdx.x;
  if (idx >= nnz4) return;
  long long i = idx >> 2;
  int h = (int)(idx & 3);
  int node = ni[i];
  alpha[idx] = alpha[idx] / (nsum[(size_t)node * 4 + h] + 1e-16f);
}

// ---------------- feature scatters (thread per (nnz, feature)) ----------------
__global__ void scatter_edge_k(const float* __restrict__ x1, const float* __restrict__ alpha,
                               const float* __restrict__ Binv, const int* __restrict__ ni,
                               const int* __restrict__ ei, float* __restrict__ out_e,
                               long long total) {
  long long idx = (long long)blockIdx.x * blockDim.x + threadIdx.x;
  if (idx >= total) return;
  long long i = idx >> 7;
  int f = (int)(idx & 127), h = f >> 5;
  int node = ni[i], edge = ei[i];
  float v = Binv[edge] * alpha[i * 4 + h] * x1[(size_t)node * IN_C + f];
  atomicAdd(&out_e[(size_t)edge * IN_C + f], v);
}

__global__ void scatter_node_k(const float* __restrict__ out_e, const float* __restrict__ alpha,
                               const float* __restrict__ Dinv, const int* __restrict__ ni,
                               const int* __restrict__ ei, float* __restrict__ out_n,
                               long long total) {
  long long idx = (long long)blockIdx.x * blockDim.x + threadIdx.x;
  if (idx >= total) return;
  long long i = idx >> 7;
  int f = (int)(idx & 127), h = f >> 5;
  int node = ni[i], edge = ei[i];
  float v = Dinv[node] * alpha[i * 4 + h] * out_e[(size_t)edge * IN_C + f];
  atomicAdd(&out_n[(size_t)node * IN_C + f], v);
}

__global__ void scatter_x2e_k(const float* __restrict__ x2, const float* __restrict__ Binv,
                              const int* __restrict__ ni, const int* __restrict__ ei,
                              float* __restrict__ oe, long long total) {
  long long idx = (long long)blockIdx.x * blockDim.x + threadIdx.x;
  if (idx >= total) return;
  long long i = idx >> 7;
  int f = (int)(idx & 127);
  int node = ni[i], edge = ei[i];
  float v = Binv[edge] * x2[(size_t)node * IN_C + f];
  atomicAdd(&oe[(size_t)edge * IN_C + f], v);
}

__global__ void scatter_e2n_k(const float* __restrict__ oe, const float* __restrict__ Dinv,
                              const int* __restrict__ ni, const int* __restrict__ ei,
                              float* __restrict__ out, long long total) {
  long long idx = (long long)blockIdx.x * blockDim.x + threadIdx.x;
  if (idx >= total) return;
  long long i = idx >> 7;
  int f = (int)(idx & 127);
  int node = ni[i], edge = ei[i];
  float v = Dinv[node] * oe[(size_t)edge * IN_C + f];
  atomicAdd(&out[(size_t)node * IN_C + f], v);
}

// ---------------- pointwise ----------------
__global__ void elu_bias_k(float* __restrict__ h, const float* __restrict__ bias1,
                           long long n) {
  long long idx = (long long)blockIdx.x * blockDim.x + threadIdx.x;
  if (idx >= n) return;
  float v = h[idx] + bias1[idx & 127];
  h[idx] = v > 0.f ? v : (expf(v) - 1.f);
}
__global__ void bias_add_k(float* __restrict__ out, const float* __restrict__ bias2,
                           long long n) {
  long long idx = (long long)blockIdx.x * blockDim.x + threadIdx.x;
  if (idx >= n) return;
  out[idx] += bias2[idx & 127];
}

// ---------------- launcher ----------------
extern "C" void kernel_launch(void* const* d_in, const int* in_sizes, int n_in,
                              void* d_out, int out_size, void* d_ws, size_t ws_size,
                              hipStream_t stream) {
  const float* x    = (const float*)d_in[0];
  const float* hea  = (const float*)d_in[1];
  const float* W1   = (const float*)d_in[2];
  const float* heW1 = (const float*)d_in[3];
  const float* att1 = (const float*)d_in[4];
  const float* b1   = (const float*)d_in[5];
  const float* W2   = (const float*)d_in[6];
  const float* b2   = (const float*)d_in[7];
  const int*   ni   = (const int*)d_in[8];
  const int*   ei   = (const int*)d_in[9];
  float* out = (float*)d_out;

  const int N   = in_sizes[0] / IN_C;
  const int M   = in_sizes[1] / IN_C;
  const int NNZ = in_sizes[8];

  float* ws = (float*)d_ws;
  size_t off = 0;
  float* x1    = ws + off; off += (size_t)N * IN_C;
  float* hef   = ws + off; off += (size_t)M * IN_C;
  size_t zoff  = off;                       // contiguous zero region begins
  float* deg   = ws + off; off += (size_t)N;            // -> Dinv in place
  float* bsz   = ws + off; off += (size_t)M;            // -> Binv in place
  float* nsum  = ws + off; off += (size_t)N * 4;
  float* out_e = ws + off; off += (size_t)M * IN_C;
  float* out_n = ws + off; off += (size_t)N * IN_C;
  size_t zcnt  = off - zoff;
  unsigned* nmax = (unsigned*)(ws + off); off += (size_t)N * 4;
  float* alpha = ws + off; off += (size_t)NNZ * 4;      // logits -> e -> alpha
  float* x2 = x1;        // x1 dead after scatter_edge
  float* oe = out_e;     // out_e dead after scatter_node (re-zeroed)

  const int B = 256;
  auto cdiv = [](long long a, long long b) { return (int)((a + b - 1) / b); };
  const long long nnz4 = (long long)NNZ * 4;
  const long long tot  = (long long)NNZ * IN_C;
  const long long nout = (long long)N * IN_C;

  // init accumulators; 0x007FFFFFu == order-preserving key of -inf
  fill_f32_k<<<2048, B, 0, stream>>>(ws + zoff, 0.f, (long long)zcnt);
  fill_f32_k<<<2048, B, 0, stream>>>(out, 0.f, nout);
  fill_u32_k<<<512, B, 0, stream>>>(nmax, 0x007FFFFFu, (long long)N * 4);

  // dense projections (WMMA f32 16x16x4)
  gemm_nt_wmma<<<cdiv(cdiv(N, 16), 8), 256, 0, stream>>>(x, W1, x1, N);
  gemm_nt_wmma<<<cdiv(cdiv(M, 16), 8), 256, 0, stream>>>(hea, heW1, hef, M);

  // degrees and reciprocals
  deg_k<<<cdiv(NNZ, B), B, 0, stream>>>(ni, ei, deg, bsz, NNZ);
  recip_k<<<cdiv(N, B), B, 0, stream>>>(deg, N);
  recip_k<<<cdiv(M, B), B, 0, stream>>>(bsz, M);

  // attention softmax (3 passes)
  attn_logits_k<<<cdiv(nnz4 * 32, B), B, 0, stream>>>(x1, hef, att1, ni, ei, alpha, nmax, NNZ);
  attn_exp_k<<<cdiv(nnz4, B), B, 0, stream>>>(ni, nmax, alpha, nsum, nnz4);
  attn_norm_k<<<cdiv(nnz4, B), B, 0, stream>>>(ni, nsum, alpha, nnz4);

  // layer 1 messages
  scatter_edge_k<<<cdiv(tot, B), B, 0, stream>>>(x1, alpha, bsz, ni, ei, out_e, tot);
  scatter_node_k<<<cdiv(tot, B), B, 0, stream>>>(out_e, alpha, deg, ni, ei, out_n, tot);
  elu_bias_k<<<cdiv(nout, B), B, 0, stream>>>(out_n, b1, nout);

  // layer 2
  gemm_nt_wmma<<<cdiv(cdiv(N, 16), 8), 256, 0, stream>>>(out_n, W2, x2, N);
  fill_f32_k<<<2048, B, 0, stream>>>(oe, 0.f, (long long)M * IN_C);
  scatter_x2e_k<<<cdiv(tot, B), B, 0, stream>>>(x2, bsz, ni, ei, oe, tot);
  scatter_e2n_k<<<cdiv(tot, B), B, 0, stream>>>(oe, deg, ni, ei, out, tot);
  bias_add_k<<<cdiv(nout, B), B, 0, stream>>>(out, b2, nout);
}